// RepAdapter_Router_2869038154038
// MI455X (gfx1250) — compile-verified
//
#include <hip/hip_runtime.h>

// ---------------------------------------------------------------------------
// RepAdapter + 2-expert router, MI455X (gfx1250, wave32).
// Memory-bound (AI ~ 4 FLOP/B): 128MB x read + 128MB out write
// => ~11us HBM floor at 23.3 TB/s. x fits the 192MB L2, so the split
// reads x from HBM once (K2) and from L2 the second time (K3).
// Both matmuls use V_WMMA_F32_16X16X4_F32 (K=4 matches H/G exactly).
// wA is zero-padded to 16 rows in workspace so the K2 inner loop has
// no divergent loads: 2x global_load_b64 + v_wmma per step, EXEC all-1s.
// ---------------------------------------------------------------------------

typedef float v2f __attribute__((ext_vector_type(2)));
typedef float v8f __attribute__((ext_vector_type(8)));

#define T_ROUTER 10.0f
#define B_   8
#define N_   1024
#define C_   4096
#define H_   8
#define G_   2
#define CG_  (C_ / G_)   // 2048
#define HG_  (H_ / G_)   // 4
#define NTOK (B_ * N_)   // 8192 tokens

// ---------------------------------------------------------------------------
// K0: zero-pad wA [8,4096] -> wApad [16,4096] (rows 8..15 = 0) so the
// WMMA B operand in K2 needs no lane predication. 256 KB, L2-resident.
// ---------------------------------------------------------------------------
__global__ __launch_bounds__(256)
void pad_wA_kernel(const float* __restrict__ wA, float* __restrict__ wApad) {
  const int i   = blockIdx.x * 256 + threadIdx.x;   // 0 .. 16*4096-1
  const int row = i >> 12;                          // /4096
  wApad[i] = (row < H_) ? wA[i] : 0.0f;
}

// ---------------------------------------------------------------------------
// K1: router logits[b,e] = dot(x[b,0,:], wE[e,:]) + bE[e]   (16 dots of 4096)
// ---------------------------------------------------------------------------
__global__ __launch_bounds__(256)
void router_logits_kernel(const float* __restrict__ x,
                          const float* __restrict__ wE,
                          const float* __restrict__ bE,
                          float* __restrict__ logits) {
  __shared__ float red[256];
  const int p = blockIdx.x;          // 0..15 -> (b,e)
  const int b = p >> 1, e = p & 1;
  const float* xr = x + (size_t)b * N_ * C_;    // token 0 of batch b
  const float* wr = wE + (size_t)e * C_;
  float acc = 0.f;
  for (int c = threadIdx.x; c < C_; c += 256)
    acc = fmaf(xr[c], wr[c], acc);
  red[threadIdx.x] = acc;
  __syncthreads();
  for (int s = 128; s > 0; s >>= 1) {
    if ((int)threadIdx.x < s) red[threadIdx.x] += red[threadIdx.x + s];
    __syncthreads();
  }
  if (threadIdx.x == 0) logits[p] = red[0] + bE[e];
}

// ---------------------------------------------------------------------------
// K2: h[t,0..7] = x[t,:] . wA^T + bA  via WMMA f32 16x16x4.
// One block = one 16-token tile; 8 waves split K=4096 into 512-wide slices
// (128 WMMAs each, 2-way accumulator ILP); partial 16x16 accumulators
// reduced through LDS. This pass is the HBM-streaming read of x.
// ---------------------------------------------------------------------------
__global__ __launch_bounds__(256)
void h_gemm_kernel(const float* __restrict__ x,
                   const float* __restrict__ wApad,   // [16,4096], rows 8+ zero
                   const float* __restrict__ bA,
                   float* __restrict__ h) {
  __shared__ float red[8][32][8];                 // 8 KB
  const int lane  = threadIdx.x & 31;
  const int wid   = threadIdx.x >> 5;
  const int t0    = blockIdx.x * 16;              // first token of tile
  const int mrow  = lane & 15;                    // A: token row / B: N column
  const int khalf = (lane >> 4) * 2;              // 0 (lanes 0-15) or 2

  // A: lane holds x[t0+mrow, k0+khalf .. +1]  (ISA 16x4 f32 A layout)
  const float* xa = x     + (size_t)(t0 + mrow) * C_ + khalf;
  // B: lane holds wApad[mrow, k0+khalf .. +1] (rows 8..15 pre-zeroed)
  const float* wb = wApad + (size_t)mrow * C_ + khalf;

  const int kbase = wid * (C_ / 8);               // 512-wide K slice
  v8f acc0 = {}, acc1 = {};
  for (int k = 0; k < C_ / 8; k += 8) {
    v2f a0 = *(const v2f*)(xa + kbase + k);
    v2f b0 = *(const v2f*)(wb + kbase + k);
    v2f a1 = *(const v2f*)(xa + kbase + k + 4);
    v2f b1 = *(const v2f*)(wb + kbase + k + 4);
    acc0 = __builtin_amdgcn_wmma_f32_16x16x4_f32(false, a0, false, b0,
                                                 (short)0, acc0, false, false);
    acc1 = __builtin_amdgcn_wmma_f32_16x16x4_f32(false, a1, false, b1,
                                                 (short)0, acc1, false, false);
  }

  #pragma unroll
  for (int v = 0; v < 8; ++v) red[wid][lane][v] = acc0[v] + acc1[v];
  __syncthreads();
  if (wid != 0) return;

  // wave 0 sums the 8 partial tiles, adds bA, writes h (D layout:
  // element (VGPR v, lane) = [token t0 + v + 8*(lane>=16), col lane&15])
  float sum[8];
  #pragma unroll
  for (int v = 0; v < 8; ++v) {
    float s = 0.f;
    #pragma unroll
    for (int w = 0; w < 8; ++w) s += red[w][lane][v];
    sum[v] = s;
  }
  const int col = lane & 15;
  if (col < H_) {
    const float bias  = bA[col];
    const int   thalf = (lane >> 4) * 8;
    #pragma unroll
    for (int v = 0; v < 8; ++v)
      h[(size_t)(t0 + v + thalf) * H_ + col] = sum[v] + bias;
  }
}

// ---------------------------------------------------------------------------
// K3: out = x + w0*(hg.wB + bB) + w1*(hg.wD + bD), one WMMA per 16x16 tile
// per matrix (K = H/G = 4 exactly). Block = 16-token tile; wave w owns
// channels [w*512, w*512+512) (a single group), so the A operand (h tile)
// is loaded once per wave and reused for all 32 channel tiles. accB/accD
// are independent -> 2-deep WMMA ILP per tile. x read hits L2 (just
// streamed by K2); out write is the second HBM pass.
// ---------------------------------------------------------------------------
__global__ __launch_bounds__(256)
void fuse_out_kernel(const float* __restrict__ x,
                     const float* __restrict__ wB,
                     const float* __restrict__ bB,
                     const float* __restrict__ wD,
                     const float* __restrict__ bD,
                     const float* __restrict__ h,
                     const float* __restrict__ logits,
                     float* __restrict__ out) {
  const int lane  = threadIdx.x & 31;
  const int wid   = threadIdx.x >> 5;
  const int t0    = blockIdx.x * 16;
  const int batch = t0 / N_;                      // uniform over the tile

  // softmax over 2 experts with temperature (cheap recompute per block)
  const float l0 = logits[batch * 2 + 0];
  const float l1 = logits[batch * 2 + 1];
  const float w0 = 1.0f / (1.0f + __expf((l1 - l0) / T_ROUTER));
  const float w1 = 1.0f - w0;

  const int n       = lane & 15;                  // N column (channel in tile)
  const int half    = lane >> 4;                  // 0 or 1
  const int chan0   = wid * (C_ / 8);             // wave's first channel
  const int g       = chan0 >> 11;                // group (2048 channels each)
  const int cw_base = chan0 - g * CG_;            // within-group base

  // A operand: h tile of group g, loaded once (lane: token t0+n, K pair)
  const v2f a = *(const v2f*)(h + (size_t)(t0 + n) * H_ + g * HG_ + half * 2);

  const float* wBg = wB + (size_t)g * CG_ * HG_;  // [2048,4], k contiguous
  const float* wDg = wD + (size_t)g * CG_ * HG_;

  for (int j = 0; j < 32; ++j) {
    const int cw0 = cw_base + j * 16;
    const int c0  = g * CG_ + cw0;                // global channel of column 0

    // B operands: 4x16 slices of the grouped-conv weights
    const v2f b2 = *(const v2f*)(wBg + (size_t)(cw0 + n) * HG_ + half * 2);
    const v2f d2 = *(const v2f*)(wDg + (size_t)(cw0 + n) * HG_ + half * 2);

    v8f accB = {}, accD = {};
    accB = __builtin_amdgcn_wmma_f32_16x16x4_f32(false, a, false, b2,
                                                 (short)0, accB, false, false);
    accD = __builtin_amdgcn_wmma_f32_16x16x4_f32(false, a, false, d2,
                                                 (short)0, accD, false, false);

    const float biasB = bB[c0 + n];
    const float biasD = bD[c0 + n];

    // D layout: (VGPR v, lane) = [token t0 + v + 8*half, channel c0 + n]
    const float* xrow = x   + (size_t)(t0 + half * 8) * C_ + c0 + n;
    float*       orow = out + (size_t)(t0 + half * 8) * C_ + c0 + n;
    #pragma unroll
    for (int v = 0; v < 8; ++v) {
      const float xv = xrow[(size_t)v * C_];
      orow[(size_t)v * C_] =
          xv + w0 * (accB[v] + biasB) + w1 * (accD[v] + biasD);
    }
  }
}

// ---------------------------------------------------------------------------
extern "C" void kernel_launch(void* const* d_in, const int* in_sizes, int n_in,
                              void* d_out, int out_size, void* d_ws,
                              size_t ws_size, hipStream_t stream) {
  (void)in_sizes; (void)n_in; (void)out_size; (void)ws_size;
  const float* x  = (const float*)d_in[0];
  const float* wA = (const float*)d_in[1];
  const float* bA = (const float*)d_in[2];
  const float* wB = (const float*)d_in[3];
  const float* bB = (const float*)d_in[4];
  const float* wD = (const float*)d_in[5];
  const float* bD = (const float*)d_in[6];
  const float* wE = (const float*)d_in[7];
  const float* bE = (const float*)d_in[8];
  float* out = (float*)d_out;

  // workspace layout (all 256B-aligned):
  float* logits = (float*)d_ws;                                   // 16 f
  float* h      = (float*)((char*)d_ws + 256);                    // 8192*8 f
  float* wApad  = (float*)((char*)d_ws + 256 + NTOK * H_ * 4);    // 16*4096 f

  pad_wA_kernel<<<dim3(16 * C_ / 256), dim3(256), 0, stream>>>(wA, wApad);
  router_logits_kernel<<<dim3(16), dim3(256), 0, stream>>>(x, wE, bE, logits);
  h_gemm_kernel<<<dim3(NTOK / 16), dim3(256), 0, stream>>>(x, wApad, bA, h);
  fuse_out_kernel<<<dim3(NTOK / 16), dim3(256), 0, stream>>>(
      x, wB, bB, wD, bD, h, logits, out);
}